// ChromaLoss_29678224015954
// MI455X (gfx1250) — compile-verified
//
#include <hip/hip_runtime.h>
#include <hip/hip_bf16.h>

// ---------------------------------------------------------------------------
// ChromaLoss: fused sRGB->Lab + CIEDE2000 + global mean on gfx1250 (wave32).
// Memory floor: 100.7MB / 23.3TB/s ~ 4.3us; kernel is TRANS-pipe bound, so all
// transcendentals are forced onto v_sin/v_cos/v_log/v_exp/v_sqrt/v_rcp/v_rsq.
// ---------------------------------------------------------------------------

#define PLANE     262144              // 512*512
#define NPIX      (16 * PLANE)        // 4194304 pixels
#define GROUPS    (NPIX / 4)          // 1048576 float4 groups
#define NBLOCKS1  2048
#define NTHREADS  256

#define SQRTF(x) __builtin_amdgcn_sqrtf(x)
#define RCPF(x)  __builtin_amdgcn_rcpf(x)
#define RSQF(x)  __builtin_amdgcn_rsqf(x)
#define EXP2F(x) exp2f(x)             // OCML exp2 == native v_exp_f32
#define DEG2RAD  0.017453292519943295f

// sRGB linearization: x<=0.04045 ? x/12.92 : ((x+0.055)/1.055)^2.4
__device__ __forceinline__ float srgb_lin(float x) {
    float y = (x + 0.055f) * (1.0f / 1.055f);
    float p = EXP2F(2.4f * __log2f(y));             // v_log_f32 + v_exp_f32
    return (x <= 0.04045f) ? x * (1.0f / 12.92f) : p;
}

// Lab f(): t>0.008856 ? cbrt(t) : 7.787t + 16/116
__device__ __forceinline__ float lab_f(float t) {
    float c = EXP2F((1.0f / 3.0f) * __log2f(t));
    return (t > 0.008856f) ? c : fmaf(7.787f, t, 16.0f / 116.0f);
}

__device__ __forceinline__ void rgb2lab(float r, float g, float b,
                                        float& L, float& A, float& B) {
    float lr = srgb_lin(r), lg = srgb_lin(g), lb = srgb_lin(b);
    float X = fmaf(0.412453f, lr, fmaf(0.357580f, lg, 0.180423f * lb)) * (1.0f / 0.95047f);
    float Y = fmaf(0.212671f, lr, fmaf(0.715160f, lg, 0.072169f * lb));
    float Z = fmaf(0.019334f, lr, fmaf(0.119193f, lg, 0.950227f * lb)) * (1.0f / 1.08883f);
    float fx = lab_f(X), fy = lab_f(Y), fz = lab_f(Z);
    L = fmaf(116.0f, fy, -16.0f);
    A = 500.0f * (fx - fy);
    B = 200.0f * (fy - fz);
}

// atan2 in degrees, result in [0, 360). Minimax poly on [0,1] + octant fixup.
__device__ __forceinline__ float hue_deg(float y, float x) {
    float ax = fabsf(x), ay = fabsf(y);
    float mx = fmaxf(ax, ay), mn = fminf(ax, ay);
    float t = mn * RCPF(mx);
    t = (mx == 0.0f) ? 0.0f : t;
    float s = t * t;
    float p = fmaf(s, fmaf(s, fmaf(s, fmaf(s, 0.0208351f, -0.085133f),
                                   0.180141f), -0.3302995f), 0.9998660f);
    float r = p * t;                                   // atan(mn/mx), [0, pi/4]
    r = (ay > ax) ? 1.5707963268f - r : r;
    r = (x < 0.0f) ? 3.1415926536f - r : r;
    float h = r * 57.295779513f;                       // degrees, [0,180]
    h = (y < 0.0f) ? -h : h;
    return (h < 0.0f) ? h + 360.0f : h;
}

__device__ __forceinline__ float pow7(float c) {
    float c2 = c * c;
    return c2 * c2 * c2 * c;
}

__device__ __forceinline__ float de2000(float L1, float a1, float b1,
                                        float L2, float a2, float b2) {
    const float P25 = 6103515625.0f;                   // 25^7
    float C1 = SQRTF(fmaf(a1, a1, b1 * b1));
    float C2 = SQRTF(fmaf(a2, a2, b2 * b2));
    float Cbar = 0.5f * (C1 + C2);
    float cb7 = pow7(Cbar);
    float G = 0.5f * (1.0f - SQRTF(cb7 * RCPF(cb7 + P25)));
    float a1p = (1.0f + G) * a1, a2p = (1.0f + G) * a2;
    float C1p = SQRTF(fmaf(a1p, a1p, b1 * b1));
    float C2p = SQRTF(fmaf(a2p, a2p, b2 * b2));
    float h1p = hue_deg(b1, a1p);
    float h2p = hue_deg(b2, a2p);
    float cc = C1p * C2p;

    float dLp = L2 - L1;
    float dCp = C2p - C1p;
    float dhp = h2p - h1p;
    dhp = (dhp > 180.0f) ? dhp - 360.0f : ((dhp < -180.0f) ? dhp + 360.0f : dhp);
    dhp = (cc == 0.0f) ? 0.0f : dhp;
    float dHp = 2.0f * SQRTF(cc) * __sinf(dhp * (0.5f * DEG2RAD));   // v_sin_f32

    float Lbp = 0.5f * (L1 + L2);
    float Cbp = 0.5f * (C1p + C2p);
    float hsum = h1p + h2p;
    float habs = fabsf(h1p - h2p);
    float hbp = (habs <= 180.0f) ? 0.5f * hsum
              : ((hsum < 360.0f) ? 0.5f * (hsum + 360.0f) : 0.5f * (hsum - 360.0f));
    hbp = (cc == 0.0f) ? hsum : hbp;

    float T = 1.0f
            - 0.17f * __cosf((hbp - 30.0f) * DEG2RAD)
            + 0.24f * __cosf((2.0f * hbp) * DEG2RAD)
            + 0.32f * __cosf(fmaf(3.0f, hbp, 6.0f) * DEG2RAD)
            - 0.20f * __cosf(fmaf(4.0f, hbp, -63.0f) * DEG2RAD);

    float u = (hbp - 275.0f) * (1.0f / 25.0f);
    float dtheta = 30.0f * EXP2F(-1.4426950408889634f * u * u);      // exp(-u^2)
    float cp7 = pow7(Cbp);
    float Rc = 2.0f * SQRTF(cp7 * RCPF(cp7 + P25));
    float l50 = Lbp - 50.0f; l50 *= l50;
    float Sl = fmaf(0.015f * l50, RSQF(20.0f + l50), 1.0f);
    float Sc = fmaf(0.045f, Cbp, 1.0f);
    float Sh = fmaf(0.015f * Cbp, T, 1.0f);
    float Rt = -__sinf((2.0f * dtheta) * DEG2RAD) * Rc;

    float tL = dLp * RCPF(Sl);
    float tC = dCp * RCPF(Sc);
    float tH = dHp * RCPF(Sh);
    float s = fmaf(tL, tL, fmaf(tC, tC, fmaf(tH, tH, Rt * tC * tH)));
    return SQRTF(fmaxf(s, 0.0f));
}

__device__ __forceinline__ float pixel_de(float r1, float g1, float b1,
                                          float r2, float g2, float b2) {
    float L1, A1, B1, L2, A2, B2;
    rgb2lab(r1, g1, b1, L1, A1, B1);
    rgb2lab(r2, g2, b2, L2, A2, B2);
    return de2000(L1, A1, B1, L2, A2, B2);
}

// Block reduction: wave32 shuffle tree -> LDS -> wave 0. Deterministic order.
__device__ __forceinline__ float block_reduce(float acc, float* smem) {
    int lane = threadIdx.x & 31;
    int wave = threadIdx.x >> 5;
    #pragma unroll
    for (int off = 16; off > 0; off >>= 1)
        acc += __shfl_down(acc, off, 32);
    if (lane == 0) smem[wave] = acc;
    __syncthreads();
    float v = 0.0f;
    if (wave == 0) {
        v = (lane < (NTHREADS / 32)) ? smem[lane] : 0.0f;
        #pragma unroll
        for (int off = 4; off > 0; off >>= 1)
            v += __shfl_down(v, off, 32);
    }
    return v;   // valid on thread 0
}

__global__ __launch_bounds__(NTHREADS)
void chroma_stage1(const float* __restrict__ img1, const float* __restrict__ img2,
                   float* __restrict__ partial) {
    __shared__ float smem[NTHREADS / 32];
    const int stride = gridDim.x * blockDim.x;
    float acc = 0.0f;
    for (int g = blockIdx.x * blockDim.x + threadIdx.x; g < GROUPS; g += stride) {
        int batch = g >> 16;                        // 65536 groups per image
        size_t base = (size_t)batch * (3 * PLANE) + (size_t)((g & 65535) << 2);

        // CDNA5: prefetch next grid-stride iteration (global_prefetch_b8)
        int gn = g + stride;
        if (gn < GROUPS) {
            size_t pn = (size_t)(gn >> 16) * (3 * PLANE) + (size_t)((gn & 65535) << 2);
            __builtin_prefetch(img1 + pn, 0, 0);
            __builtin_prefetch(img1 + pn + PLANE, 0, 0);
            __builtin_prefetch(img1 + pn + 2 * PLANE, 0, 0);
            __builtin_prefetch(img2 + pn, 0, 0);
            __builtin_prefetch(img2 + pn + PLANE, 0, 0);
            __builtin_prefetch(img2 + pn + 2 * PLANE, 0, 0);
        }

        // 6 x global_load_b128: 4 pixels, both images, all 3 channel planes
        float4 r1 = *(const float4*)(img1 + base);
        float4 g1 = *(const float4*)(img1 + base + PLANE);
        float4 b1 = *(const float4*)(img1 + base + 2 * PLANE);
        float4 r2 = *(const float4*)(img2 + base);
        float4 g2 = *(const float4*)(img2 + base + PLANE);
        float4 b2 = *(const float4*)(img2 + base + 2 * PLANE);

        acc += pixel_de(r1.x, g1.x, b1.x, r2.x, g2.x, b2.x);
        acc += pixel_de(r1.y, g1.y, b1.y, r2.y, g2.y, b2.y);
        acc += pixel_de(r1.z, g1.z, b1.z, r2.z, g2.z, b2.z);
        acc += pixel_de(r1.w, g1.w, b1.w, r2.w, g2.w, b2.w);
    }
    float v = block_reduce(acc, smem);
    if (threadIdx.x == 0) partial[blockIdx.x] = v;
}

__global__ __launch_bounds__(NTHREADS)
void chroma_stage2(const float* __restrict__ partial, float* __restrict__ out, int n) {
    __shared__ float smem[NTHREADS / 32];
    float acc = 0.0f;
    for (int i = threadIdx.x; i < n; i += NTHREADS) acc += partial[i];
    float v = block_reduce(acc, smem);
    if (threadIdx.x == 0) out[0] = v * (1.0f / (float)NPIX);
}

extern "C" void kernel_launch(void* const* d_in, const int* in_sizes, int n_in,
                              void* d_out, int out_size, void* d_ws, size_t ws_size,
                              hipStream_t stream) {
    const float* img1 = (const float*)d_in[0];
    const float* img2 = (const float*)d_in[1];
    float* partial = (float*)d_ws;                 // NBLOCKS1 floats (8 KB)
    chroma_stage1<<<NBLOCKS1, NTHREADS, 0, stream>>>(img1, img2, partial);
    chroma_stage2<<<1, NTHREADS, 0, stream>>>(partial, (float*)d_out, NBLOCKS1);
}